// RatioBoxGroupProjector_1838246003111
// MI455X (gfx1250) — compile-verified
//
#include <hip/hip_runtime.h>

// ---------------- problem constants (match reference) ----------------
constexpr int   NG        = 1024;   // n_groups (fixed in reference)
constexpr int   MAX_ITERS = 30;
constexpr float TAU       = 0.1f;
constexpr float GAMMA     = 0.05f;

constexpr int   NBLK = 1024;        // blocks for streaming kernels
constexpr int   TPB  = 256;         // 8 wave32 waves / workgroup on CDNA5

typedef unsigned int u32x4 __attribute__((ext_vector_type(4)));
typedef int          i32x8 __attribute__((ext_vector_type(8)));
typedef int          i32x4 __attribute__((ext_vector_type(4)));

__device__ __forceinline__ float clip09(float x) {
  // w*clip(y,l,u) == clip(w*y, 1-TAU, 1+TAU) since w*l = 1-TAU, w*u = 1+TAU exactly
  return fminf(fmaxf(x, 1.0f - TAU), 1.0f + TAU);
}

// ---------------- K0: init per-group stat accumulators ----------------
__global__ void k0_init(float* ngc, float* s0, unsigned* tminb, unsigned* tmaxb) {
  int g = blockIdx.x * blockDim.x + threadIdx.x;
  if (g < NG) {
    ngc[g]   = 0.0f;
    s0[g]    = 0.0f;
    tminb[g] = 0x7F800000u;  // +inf (t >= 0, float bits are order-monotone)
    tmaxb[g] = 0u;
  }
}

// ---------------- K1: t = y_raw/max(y_real,eps); per-group count/S0/min/max ----
__global__ void k1_stats(const float* __restrict__ yraw, const float* __restrict__ yreal,
                         const int* __restrict__ gid, float* __restrict__ t,
                         float* __restrict__ ngc, float* __restrict__ s0,
                         unsigned* __restrict__ tminb, unsigned* __restrict__ tmaxb,
                         int n) {
  __shared__ float    scnt[NG];
  __shared__ float    ssum[NG];
  __shared__ unsigned smn[NG];
  __shared__ unsigned smx[NG];
  for (int j = threadIdx.x; j < NG; j += blockDim.x) {
    scnt[j] = 0.0f; ssum[j] = 0.0f; smn[j] = 0x7F800000u; smx[j] = 0u;
  }
  __syncthreads();

  const int tid    = blockIdx.x * blockDim.x + threadIdx.x;
  const int stride = gridDim.x * blockDim.x;
  const int nvec   = n >> 2;
  const float4* a4 = (const float4*)yraw;
  const float4* b4 = (const float4*)yreal;
  const int4*   g4 = (const int4*)gid;
  float4*       t4 = (float4*)t;

  for (int c = tid; c < nvec; c += stride) {
    float4 a = a4[c]; float4 b = b4[c]; int4 g = g4[c];
    float4 tv;
    {
      float yre = fmaxf(b.x, 1e-9f); float ti = a.x * (1.0f / yre); tv.x = ti;
      atomicAdd(&scnt[g.x], 1.0f); atomicAdd(&ssum[g.x], clip09(ti));
      atomicMin(&smn[g.x], __float_as_uint(ti)); atomicMax(&smx[g.x], __float_as_uint(ti));
    }
    {
      float yre = fmaxf(b.y, 1e-9f); float ti = a.y * (1.0f / yre); tv.y = ti;
      atomicAdd(&scnt[g.y], 1.0f); atomicAdd(&ssum[g.y], clip09(ti));
      atomicMin(&smn[g.y], __float_as_uint(ti)); atomicMax(&smx[g.y], __float_as_uint(ti));
    }
    {
      float yre = fmaxf(b.z, 1e-9f); float ti = a.z * (1.0f / yre); tv.z = ti;
      atomicAdd(&scnt[g.z], 1.0f); atomicAdd(&ssum[g.z], clip09(ti));
      atomicMin(&smn[g.z], __float_as_uint(ti)); atomicMax(&smx[g.z], __float_as_uint(ti));
    }
    {
      float yre = fmaxf(b.w, 1e-9f); float ti = a.w * (1.0f / yre); tv.w = ti;
      atomicAdd(&scnt[g.w], 1.0f); atomicAdd(&ssum[g.w], clip09(ti));
      atomicMin(&smn[g.w], __float_as_uint(ti)); atomicMax(&smx[g.w], __float_as_uint(ti));
    }
    t4[c] = tv;
  }
  for (int i = (nvec << 2) + tid; i < n; i += stride) {  // scalar tail
    float yre = fmaxf(yreal[i], 1e-9f); float ti = yraw[i] * (1.0f / yre); t[i] = ti;
    int g = gid[i];
    atomicAdd(&scnt[g], 1.0f); atomicAdd(&ssum[g], clip09(ti));
    atomicMin(&smn[g], __float_as_uint(ti)); atomicMax(&smx[g], __float_as_uint(ti));
  }
  __syncthreads();
  for (int j = threadIdx.x; j < NG; j += blockDim.x) {
    if (scnt[j] != 0.0f) {
      atomicAdd(&ngc[j], scnt[j]);
      atomicAdd(&s0[j],  ssum[j]);
      atomicMin(&tminb[j], smn[j]);
      atomicMax(&tmaxb[j], smx[j]);
    }
  }
}

// ---------------- K2: per-group feasibility, bracket, first mid ----------------
__global__ void k2_ginit(const float* ngc, const float* s0, const unsigned* tminb,
                         const unsigned* tmaxb, float* lo, float* hi, float* midv,
                         float* Tg, unsigned* feas) {
  int g = blockIdx.x * blockDim.x + threadIdx.x;
  if (g >= NG) return;
  float n = ngc[g];
  float L = (1.0f - GAMMA) * n;
  float U = (1.0f + GAMMA) * n;
  float s = s0[g];
  unsigned fe = (s >= L && s <= U) ? 1u : 0u;
  float tmax = __uint_as_float(tmaxb[g]);
  float tmin = __uint_as_float(tminb[g]);
  // (l - y_raw)*w = (1-TAU) - t ; (u - y_raw)*w = (1+TAU) - t
  float alo = ((1.0f - TAU) - tmax) - 1.0f;
  float ahi = ((1.0f + TAU) - tmin) + 1.0f;
  if (!(n > 0.0f)) { alo = 0.0f; ahi = 0.0f; }   // empty-group guard
  lo[g]   = alo;
  hi[g]   = ahi;
  midv[g] = 0.5f * (alo + ahi);                  // mid_1
  Tg[g]   = (s < L) ? L : U;
  feas[g] = fe;
}

// ---------------- K3: one bisection sweep: partial[b][g] = sum clip(t+mid[g]) ----
__global__ void k3_sum(const float* __restrict__ t, const int* __restrict__ gid,
                       const float* __restrict__ midv, float* __restrict__ partial,
                       int n) {
  // sb[0..NG)   : mid table  (TDM destination, LDS byte offset 0)
  // sb[NG..2NG) : per-group accumulation bins
  __shared__ float sb[2 * NG];

  if (threadIdx.x == 0) {
    // Tensor Data Mover: DMA the 4KB mid table into LDS (1-D tile, 4B elems).
    // D# per CDNA5 ISA; groups 2/3 unused (<=2D).
    unsigned long long ga = (unsigned long long)midv;
    u32x4 g0;
    g0.x = 1u;                                        // count=1, user mode
    g0.y = 0u;                                        // lds_addr = 0 (sb base)
    g0.z = (unsigned)(ga & 0xFFFFFFFFull);            // global_addr[31:0]
    g0.w = (unsigned)((ga >> 32) & 0x01FFFFFFull)     // global_addr[56:32]
         | 0x80000000u;                               // type = 2 ("image")
    i32x8 g1;
    g1[0] = 0x00020000;        // workgroup_mask=0, data_size=2 -> 4 bytes
    g1[1] = (int)(NG << 16);   // tensor_dim0 = 1024 (bits 63:48)
    g1[2] = (int)(1  << 16);   // tensor_dim0 hi = 0, tensor_dim1 = 1
    g1[3] = (int)(NG << 16);   // tensor_dim1 hi = 0, tile_dim0 = 1024
    g1[4] = 1;                 // tile_dim1 = 1, tile_dim2 = 0
    g1[5] = NG;                // tensor_dim0_stride = 1024
    g1[6] = 0;
    g1[7] = 0;
    i32x4 gz4 = {0, 0, 0, 0};           // groups 2/3 unused (<=2D tile)
    i32x8 gz8 = {0, 0, 0, 0, 0, 0, 0, 0};
    __builtin_amdgcn_tensor_load_to_lds(g0, g1, gz4, gz4, gz8, 0);
    __builtin_amdgcn_s_wait_tensorcnt(0);
  }
  for (int j = threadIdx.x; j < NG; j += blockDim.x) sb[NG + j] = 0.0f;
  __syncthreads();

  const int tid    = blockIdx.x * blockDim.x + threadIdx.x;
  const int stride = gridDim.x * blockDim.x;
  const int nvec   = n >> 2;
  const float4* t4 = (const float4*)t;
  const int4*   g4 = (const int4*)gid;

  for (int c = tid; c < nvec; c += stride) {
    __builtin_prefetch(&t4[c + stride], 0, 1);   // global_prefetch_b8
    float4 tv = t4[c]; int4 gv = g4[c];
    atomicAdd(&sb[NG + gv.x], clip09(tv.x + sb[gv.x]));
    atomicAdd(&sb[NG + gv.y], clip09(tv.y + sb[gv.y]));
    atomicAdd(&sb[NG + gv.z], clip09(tv.z + sb[gv.z]));
    atomicAdd(&sb[NG + gv.w], clip09(tv.w + sb[gv.w]));
  }
  for (int i = (nvec << 2) + tid; i < n; i += stride) {  // tail
    int g = gid[i];
    atomicAdd(&sb[NG + g], clip09(t[i] + sb[g]));
  }
  __syncthreads();
  for (int j = threadIdx.x; j < NG; j += blockDim.x)
    partial[(size_t)blockIdx.x * NG + j] = sb[NG + j];   // coalesced, deterministic layout
}

// ---------------- K4: per-group reduce of partials + bisection update ----------
__global__ void k4_upd(const float* __restrict__ partial, const float* __restrict__ Tg,
                       float* __restrict__ lo, float* __restrict__ hi,
                       float* __restrict__ midv, int nblk) {
  __shared__ float red[TPB];
  const int g = blockIdx.x;                 // one block per group
  float s = 0.0f;
  for (int b = threadIdx.x; b < nblk; b += TPB)
    s += partial[(size_t)b * NG + g];
  red[threadIdx.x] = s;
  __syncthreads();
  for (int off = TPB >> 1; off > 0; off >>= 1) {
    if (threadIdx.x < off) red[threadIdx.x] += red[threadIdx.x + off];
    __syncthreads();
  }
  if (threadIdx.x == 0) {
    float Smid = red[0];
    float m = midv[g];
    float l = lo[g], h = hi[g];
    if (Smid < Tg[g]) l = m; else h = m;
    lo[g] = l; hi[g] = h;
    midv[g] = 0.5f * (l + h);               // next iteration's mid
  }
}

// ---------------- K5: final select/output ----------------
__global__ void k5_out(const float* __restrict__ yraw, const float* __restrict__ yreal,
                       const int* __restrict__ gid, const float* __restrict__ midv,
                       const unsigned* __restrict__ feas, float* __restrict__ out,
                       int n) {
  __shared__ float    smid[NG];
  __shared__ unsigned sfe[NG];
  for (int j = threadIdx.x; j < NG; j += blockDim.x) { smid[j] = midv[j]; sfe[j] = feas[j]; }
  __syncthreads();

  const int tid    = blockIdx.x * blockDim.x + threadIdx.x;
  const int stride = gridDim.x * blockDim.x;
  const int nvec   = n >> 2;
  const float4* a4 = (const float4*)yraw;
  const float4* b4 = (const float4*)yreal;
  const int4*   g4 = (const int4*)gid;
  float4*       o4 = (float4*)out;

  for (int c = tid; c < nvec; c += stride) {
    float4 a = a4[c]; float4 b = b4[c]; int4 g = g4[c];
    float4 r;
    {
      float yre = fmaxf(b.x, 1e-9f); float l = (1.0f - TAU) * yre, u = (1.0f + TAU) * yre;
      float yc = fminf(fmaxf(a.x, l), u);
      float yb = fminf(fmaxf(a.x + smid[g.x] * yre, l), u);   // mid/w = mid*yre
      r.x = sfe[g.x] ? yc : yb;
    }
    {
      float yre = fmaxf(b.y, 1e-9f); float l = (1.0f - TAU) * yre, u = (1.0f + TAU) * yre;
      float yc = fminf(fmaxf(a.y, l), u);
      float yb = fminf(fmaxf(a.y + smid[g.y] * yre, l), u);
      r.y = sfe[g.y] ? yc : yb;
    }
    {
      float yre = fmaxf(b.z, 1e-9f); float l = (1.0f - TAU) * yre, u = (1.0f + TAU) * yre;
      float yc = fminf(fmaxf(a.z, l), u);
      float yb = fminf(fmaxf(a.z + smid[g.z] * yre, l), u);
      r.z = sfe[g.z] ? yc : yb;
    }
    {
      float yre = fmaxf(b.w, 1e-9f); float l = (1.0f - TAU) * yre, u = (1.0f + TAU) * yre;
      float yc = fminf(fmaxf(a.w, l), u);
      float yb = fminf(fmaxf(a.w + smid[g.w] * yre, l), u);
      r.w = sfe[g.w] ? yc : yb;
    }
    o4[c] = r;
  }
  for (int i = (nvec << 2) + tid; i < n; i += stride) {   // tail
    float yre = fmaxf(yreal[i], 1e-9f);
    float l = (1.0f - TAU) * yre, u = (1.0f + TAU) * yre;
    float yc = fminf(fmaxf(yraw[i], l), u);
    float yb = fminf(fmaxf(yraw[i] + smid[gid[i]] * yre, l), u);
    out[i] = sfe[gid[i]] ? yc : yb;
  }
}

// ---------------- launch ----------------
extern "C" void kernel_launch(void* const* d_in, const int* in_sizes, int n_in,
                              void* d_out, int out_size, void* d_ws, size_t ws_size,
                              hipStream_t stream) {
  (void)n_in; (void)out_size; (void)ws_size;
  const float* yraw  = (const float*)d_in[0];
  const float* yreal = (const float*)d_in[1];
  const int*   gid   = (const int*)d_in[2];
  const int n = in_sizes[0];
  float* out = (float*)d_out;

  // workspace carve (all 4B types; t at offset 0 is 16B aligned for float4)
  float*    t     = (float*)d_ws;
  float*    part  = t + (size_t)n;
  float*    ngc   = part + (size_t)NBLK * NG;
  float*    s0    = ngc + NG;
  unsigned* tminb = (unsigned*)(s0 + NG);
  unsigned* tmaxb = tminb + NG;
  float*    lo    = (float*)(tmaxb + NG);
  float*    hi    = lo + NG;
  float*    midv  = hi + NG;
  float*    Tg    = midv + NG;
  unsigned* feas  = (unsigned*)(Tg + NG);

  dim3 big(NBLK), blk(TPB), small((NG + TPB - 1) / TPB);

  k0_init<<<small, blk, 0, stream>>>(ngc, s0, tminb, tmaxb);
  k1_stats<<<big, blk, 0, stream>>>(yraw, yreal, gid, t, ngc, s0, tminb, tmaxb, n);
  k2_ginit<<<small, blk, 0, stream>>>(ngc, s0, tminb, tmaxb, lo, hi, midv, Tg, feas);

  // reference scan: mid of step 30 is what the output uses -> 29 [sweep,update]
  for (int k = 0; k < MAX_ITERS - 1; ++k) {
    k3_sum<<<big, blk, 0, stream>>>(t, gid, midv, part, n);
    k4_upd<<<dim3(NG), blk, 0, stream>>>(part, Tg, lo, hi, midv, NBLK);
  }

  k5_out<<<big, blk, 0, stream>>>(yraw, yreal, gid, midv, feas, out, n);
}